// ContrastiveLossWithConv_52888227283406
// MI455X (gfx1250) — compile-verified
//
#include <hip/hip_runtime.h>
#include <math.h>

typedef __attribute__((ext_vector_type(2))) float v2f;
typedef __attribute__((ext_vector_type(8))) float v8f;

#define NBATCH 8
#define NCH    32
#define IMH    128
#define IMW    128
#define TILEW  16          // output tile width  (pixels along w per wave row)
#define TILEH  8           // output tile height (one row per wave, 8 waves)
#define HALOW  18          // TILEW + 2
#define HALOH  10          // TILEH + 2
#define CPAD   34          // channel stride in LDS: even + conflict-free across 16 lanes
#define NBLK   (NBATCH * (IMH / TILEH) * (IMW / TILEW))   // 8*16*8 = 1024
#define NPIX   (NBATCH * IMH * IMW)                       // 131072
#define EPSN   1e-12f

#if defined(__HIP_DEVICE_COMPILE__) && \
    __has_builtin(__builtin_amdgcn_global_load_async_to_lds_b32) && \
    __has_builtin(__builtin_amdgcn_s_wait_asynccnt)
#define USE_ASYNC_LDS 1
typedef __attribute__((address_space(1))) int as1int;
typedef __attribute__((address_space(3))) int as3int;
#else
#define USE_ASYNC_LDS 0
#endif

__global__ __launch_bounds__(256)
void ContrastiveLossWithConv_main(const float* __restrict__ A,
                                  const float* __restrict__ B,
                                  float* __restrict__ partial) {
    __shared__ float sB[HALOH * HALOW * CPAD];   // ~24.5 KB (WGP has 320 KB)
    __shared__ float sRed[256];

    const int tid = threadIdx.x;
    const int blk = blockIdx.x;
    const int ntw = IMW / TILEW;                 // 8
    const int nth = IMH / TILEH;                 // 16
    const int tw  = blk % ntw;
    const int th  = (blk / ntw) % nth;
    const int bb  = blk / (ntw * nth);
    const int w0  = tw * TILEW;
    const int h0  = th * TILEH;

    const float* __restrict__ Ab = A + (size_t)bb * NCH * IMH * IMW;
    const float* __restrict__ Bb = B + (size_t)bb * NCH * IMH * IMW;

    // ---- Stage raw B tile (+halo, zero-padded) into LDS -------------------
    for (int e = tid; e < NCH * HALOH * HALOW; e += 256) {
        const int x = e % HALOW;
        const int r = e / HALOW;
        const int y = r % HALOH;
        const int c = r / HALOH;
        const int gy = h0 - 1 + y;
        const int gx = w0 - 1 + x;
        const int lidx = (y * HALOW + x) * CPAD + c;
        const bool inb = (gy >= 0) & (gy < IMH) & (gx >= 0) & (gx < IMW);
#if USE_ASYNC_LDS
        if (inb) {
            __builtin_amdgcn_global_load_async_to_lds_b32(
                (as1int*)(Bb + ((size_t)c * IMH + gy) * IMW + gx),
                (as3int*)&sB[lidx], 0, 0);
        } else {
            sB[lidx] = 0.0f;
        }
#else
        sB[lidx] = inb ? Bb[((size_t)c * IMH + gy) * IMW + gx] : 0.0f;
#endif
    }
#if USE_ASYNC_LDS
    __builtin_amdgcn_s_wait_asynccnt(0);
#endif
    __syncthreads();

    // ---- L2-normalize each staged B pixel in place ------------------------
    for (int p = tid; p < HALOH * HALOW; p += 256) {
        float* px = &sB[p * CPAD];
        float s = 0.0f;
#pragma unroll
        for (int c = 0; c < NCH; ++c) { const float v = px[c]; s += v * v; }
        const float inv = 1.0f / fmaxf(sqrtf(s), EPSN);
#pragma unroll
        for (int c = 0; c < NCH; ++c) px[c] *= inv;
    }
    __syncthreads();

    // ---- Per-wave: one output row of 16 pixels ----------------------------
    const int lane = tid & 31;
    const int wave = tid >> 5;        // output row within tile
    const int pcol = lane & 15;       // pixel column (M and N index)
    const int half = lane >> 4;       // which K-half this lane supplies
    const int h    = h0 + wave;
    const int w    = w0 + pcol;

    // Load A operand chunks (channels 4q + 2*half + {0,1}) and normalize.
    v2f an[8];
    float nrm = 0.0f;
#pragma unroll
    for (int q = 0; q < 8; ++q) {
        const int c = 4 * q + 2 * half;
        const float a0 = Ab[((size_t)c * IMH + h) * IMW + w];
        const float a1 = Ab[((size_t)(c + 1) * IMH + h) * IMW + w];
        an[q][0] = a0; an[q][1] = a1;
        nrm += a0 * a0 + a1 * a1;
    }
    nrm += __shfl_xor(nrm, 16);       // combine the two K-halves (wave32)
    const float invA = 1.0f / fmaxf(sqrtf(nrm), EPSN);
#pragma unroll
    for (int q = 0; q < 8; ++q) { an[q][0] *= invA; an[q][1] *= invA; }

    // sim_k via D = An(16x32) x Bn_shift(32x16); diagonal = per-pixel dots.
    float sims[9];
#pragma unroll
    for (int k = 0; k < 9; ++k) {
        const int di = k / 3 - 1;
        const int dj = k % 3 - 1;
        const float* bp = &sB[((wave + 1 + di) * HALOW + (pcol + 1 + dj)) * CPAD];
        v8f acc = {};
#pragma unroll
        for (int q = 0; q < 8; ++q) {
            const int c = 4 * q + 2 * half;
            v2f bn; bn[0] = bp[c]; bn[1] = bp[c + 1];
            acc = __builtin_amdgcn_wmma_f32_16x16x4_f32(
                false, an[q], false, bn, (short)0, acc, false, false);
        }
        // Diagonal (p,p): element p&7 on lane p (p<8) / lane p+16 (p>=8).
        const int e = lane & 7;
        float v = acc[0];
#pragma unroll
        for (int j = 1; j < 8; ++j) v = (e == j) ? acc[j] : v;
        sims[k] = v;
    }

    // Lanes {0..7, 24..31} own their pixel's diagonal locally; others idle.
    float part = 0.0f;
    if ((lane < 8) | (lane >= 24)) {
        float z[9], zmax = -1e30f;
#pragma unroll
        for (int k = 0; k < 9; ++k) { z[k] = sims[k] / 0.07f; zmax = fmaxf(zmax, z[k]); }
        float se = 0.0f;
#pragma unroll
        for (int k = 0; k < 9; ++k) se += expf(z[k] - zmax);
        const float lse = zmax + logf(se);
#pragma unroll
        for (int k = 0; k < 9; ++k)
            if (sims[k] > 0.5f) part += (z[k] - lse);   // labels * log_prob
    }

    // ---- Workgroup reduction → one partial per block ----------------------
    sRed[tid] = part;
    __syncthreads();
    for (int off = 128; off > 0; off >>= 1) {
        if (tid < off) sRed[tid] += sRed[tid + off];
        __syncthreads();
    }
    if (tid == 0) partial[blk] = sRed[0];
}

__global__ __launch_bounds__(256)
void ContrastiveLossWithConv_reduce(const float* __restrict__ partial,
                                    float* __restrict__ out) {
    __shared__ float r[256];
    float s = 0.0f;
    for (int i = threadIdx.x; i < NBLK; i += 256) s += partial[i];
    r[threadIdx.x] = s;
    __syncthreads();
    for (int off = 128; off > 0; off >>= 1) {
        if (threadIdx.x < off) r[threadIdx.x] += r[threadIdx.x + off];
        __syncthreads();
    }
    if (threadIdx.x == 0) out[0] = -r[0] * (1.0f / (float)NPIX);
}

extern "C" void kernel_launch(void* const* d_in, const int* in_sizes, int n_in,
                              void* d_out, int out_size, void* d_ws, size_t ws_size,
                              hipStream_t stream) {
    const float* A = (const float*)d_in[0];
    const float* B = (const float*)d_in[1];
    float* out     = (float*)d_out;
    float* partial = (float*)d_ws;   // NBLK floats = 4 KB scratch

    ContrastiveLossWithConv_main<<<dim3(NBLK), dim3(256), 0, stream>>>(A, B, partial);
    ContrastiveLossWithConv_reduce<<<dim3(1), dim3(256), 0, stream>>>(partial, out);
}